// ViterbiLoss_30279519437537
// MI455X (gfx1250) — compile-verified
//
#include <hip/hip_runtime.h>
#include <hip/hip_bf16.h>

#define TAGSET    48
#define START_TAG 46
#define END_TAG   47
#define B_        32
#define T_        512
#define KK        (48 * 48)   /* 2304 */
#define CHUNKS    16
#define CLEN      (T_ / CHUNKS)  /* 32 */
#define NTHR      288            /* 9 waves of 32: 3x3 tiles of 16x16 */

typedef float v2f __attribute__((ext_vector_type(2)));
typedef float v8f __attribute__((ext_vector_type(8)));

// ---------------------------------------------------------------------------
// Phase 1: per (batch, chunk) compose transition matrices in exp-domain:
//   P_hat <- P_hat @ exp(s_t - m_t),  sigma += m_t   (48x48 f32 matmul/step)
// using V_WMMA_F32_16X16X4_F32 on 3x3 tiles of 16x16, 12 k-tiles per step.
// Global loads for step t+1 are issued before the WMMA of step t (software
// pipeline); renormalization is done from the D-tile registers.
// ---------------------------------------------------------------------------
__global__ __launch_bounds__(NTHR) void crf_chunk_kernel(
    const float* __restrict__ scores, const int* __restrict__ lengths,
    float* __restrict__ wsP, float* __restrict__ wsSigma)
{
    __shared__ float Pbuf[2][KK];   // double-buffered running product
    __shared__ float ET[KK];        // exp(s_t - m) stored TRANSPOSED: ET[j*48+i]
    __shared__ float sWave[16];
    __shared__ float sBroadcast;

    const int tid  = threadIdx.x;
    const int lane = tid & 31;
    const int wid  = tid >> 5;
    const int blk  = blockIdx.x;
    const int b    = blk / CHUNKS;
    const int c    = blk % CHUNKS;

    // P = Identity
    for (int i = tid; i < KK; i += NTHR)
        Pbuf[0][i] = ((i / 48) == (i % 48)) ? 1.0f : 0.0f;

    const int len = lengths[b];
    int tStart = c * CLEN; if (tStart < 1) tStart = 1;       // recurrence starts at t=1
    int tEnd   = (c + 1) * CLEN; if (tEnd > len) tEnd = len; // valid mask: t < len

    const int tileI = wid / 3, tileJ = wid % 3;
    const int l16   = lane & 15, lh = lane >> 4;
    const int aRow  = tileI * 16 + l16;   // A-fragment row (M)
    const int bCol  = tileJ * 16 + l16;   // B-fragment col (N)

    float sigma = 0.0f;
    int   cur   = 0;

    // ---- prologue: load s_{tStart} into registers ----
    float4 r0 = make_float4(0.f, 0.f, 0.f, 0.f);
    float4 r1 = r0;
    if (tStart < tEnd) {
        const float* sp = scores + (size_t)(b * T_ + tStart) * KK;
        r0 = ((const float4*)sp)[tid];
        r1 = ((const float4*)sp)[tid + NTHR];
    }

    for (int t = tStart; t < tEnd; ++t) {
        // ---- block max m_t over current registers ----
        float lm = fmaxf(fmaxf(fmaxf(r0.x, r0.y), fmaxf(r0.z, r0.w)),
                         fmaxf(fmaxf(r1.x, r1.y), fmaxf(r1.z, r1.w)));
        #pragma unroll
        for (int off = 16; off > 0; off >>= 1)
            lm = fmaxf(lm, __shfl_down(lm, off, 32));
        if (lane == 0) sWave[wid] = lm;
        __syncthreads();                      // also fences prior-iter ET readers
        if (tid == 0) {
            float m = sWave[0];
            for (int w = 1; w < 9; ++w) m = fmaxf(m, sWave[w]);
            sBroadcast = m;
        }
        __syncthreads();
        const float m = sBroadcast;
        sigma += m;

        // ---- E = exp(s - m), stored transposed for contiguous B fragments ----
        {
            int f0 = 4 * tid;            // 48 % 4 == 0 -> no row wrap in a group
            int i0 = f0 / 48, j0 = f0 % 48;
            ET[(j0 + 0) * 48 + i0] = __expf(r0.x - m);
            ET[(j0 + 1) * 48 + i0] = __expf(r0.y - m);
            ET[(j0 + 2) * 48 + i0] = __expf(r0.z - m);
            ET[(j0 + 3) * 48 + i0] = __expf(r0.w - m);
            int f1 = f0 + 4 * NTHR;
            int i1 = f1 / 48, j1 = f1 % 48;
            ET[(j1 + 0) * 48 + i1] = __expf(r1.x - m);
            ET[(j1 + 1) * 48 + i1] = __expf(r1.y - m);
            ET[(j1 + 2) * 48 + i1] = __expf(r1.z - m);
            ET[(j1 + 3) * 48 + i1] = __expf(r1.w - m);
        }

        // ---- prefetch s_{t+1}: overlaps the barrier + WMMA below ----
        {
            const int tn = (t + 1 < tEnd) ? t + 1 : t;   // clamped, always valid
            const float* spn = scores + (size_t)(b * T_ + tn) * KK;
            r0 = ((const float4*)spn)[tid];
            r1 = ((const float4*)spn)[tid + NTHR];
        }
        __syncthreads();

        // ---- Pnew = Pold @ E via v_wmma_f32_16x16x4_f32 ----
        const float* P0 = Pbuf[cur];
        float*       P1 = Pbuf[cur ^ 1];
        v8f acc = {};
        #pragma unroll
        for (int kt = 0; kt < 12; ++kt) {
            const int k = kt * 4 + 2 * lh;            // K half-split at lane 16
            v2f a, bf;
            a.x  = P0[aRow * 48 + k];                  // A: M=l%16, K={k,k+1}
            a.y  = P0[aRow * 48 + k + 1];
            bf.x = ET[bCol * 48 + k];                  // B: N=l%16, K={k,k+1}
            bf.y = ET[bCol * 48 + k + 1];
            acc = __builtin_amdgcn_wmma_f32_16x16x4_f32(
                      false, a, false, bf, (short)0, acc, false, false);
        }
        cur ^= 1;

        // ---- periodic renormalization straight from the D-tile registers ----
        float scale = 1.0f;
        if (((t - tStart) & 7) == 7) {     // uniform across block
            float pm = 0.0f;               // entries are non-negative
            #pragma unroll
            for (int v = 0; v < 8; ++v) pm = fmaxf(pm, acc[v]);
            #pragma unroll
            for (int off = 16; off > 0; off >>= 1)
                pm = fmaxf(pm, __shfl_down(pm, off, 32));
            if (lane == 0) sWave[wid] = pm;
            __syncthreads();
            if (tid == 0) {
                float mm = sWave[0];
                for (int w = 1; w < 9; ++w) mm = fmaxf(mm, sWave[w]);
                sBroadcast = mm;
            }
            __syncthreads();
            const float pmax = sBroadcast;
            scale = 1.0f / pmax;
            sigma += __logf(pmax);
        }

        #pragma unroll
        for (int v = 0; v < 8; ++v) {      // D: M=v+8*(lane/16), N=l%16
            int r = tileI * 16 + v + 8 * lh;
            P1[r * 48 + bCol] = acc[v] * scale;
        }
    }

    __syncthreads();
    float* outP = wsP + (size_t)blk * KK;
    for (int i = tid; i < KK; i += NTHR) outP[i] = Pbuf[cur][i];
    if (tid == 0) wsSigma[blk] = sigma;
}

// ---------------------------------------------------------------------------
// Phase 2: fold alpha0 through the 16 chunk matrices (per batch).
// ---------------------------------------------------------------------------
__global__ __launch_bounds__(64) void crf_combine_kernel(
    const float* __restrict__ scores,
    const float* __restrict__ wsP, const float* __restrict__ wsSigma,
    float* __restrict__ wsAlphaEnd)
{
    __shared__ float v[48];
    __shared__ float red[64];
    const int b = blockIdx.x, tid = threadIdx.x;

    float a0 = -__FLT_MAX__;
    if (tid < 48) a0 = scores[(size_t)(b * T_) * KK + START_TAG * 48 + tid];
    red[tid] = a0;
    __syncthreads();
    #pragma unroll
    for (int off = 32; off > 0; off >>= 1) {
        if (tid < off) red[tid] = fmaxf(red[tid], red[tid + off]);
        __syncthreads();
    }
    const float m0 = red[0];
    __syncthreads();
    float sigma = m0;
    if (tid < 48) v[tid] = __expf(a0 - m0);
    __syncthreads();

    for (int c = 0; c < CHUNKS; ++c) {
        const float* P = wsP + (size_t)(b * CHUNKS + c) * KK;
        float nv = 0.0f;
        if (tid < 48)
            for (int i = 0; i < 48; ++i) nv += v[i] * P[i * 48 + tid];
        red[tid] = (tid < 48) ? nv : 0.0f;   // values are non-negative
        __syncthreads();
        #pragma unroll
        for (int off = 32; off > 0; off >>= 1) {
            if (tid < off) red[tid] = fmaxf(red[tid], red[tid + off]);
            __syncthreads();
        }
        const float vm = red[0];
        __syncthreads();
        if (tid < 48) v[tid] = nv / vm;
        sigma += wsSigma[b * CHUNKS + c] + __logf(vm);
        __syncthreads();
    }
    if (tid == END_TAG) wsAlphaEnd[b] = sigma + __logf(v[END_TAG]);
}

// ---------------------------------------------------------------------------
// Gold-path score: masked gather-sum, deterministic tree reduction per batch.
// ---------------------------------------------------------------------------
__global__ __launch_bounds__(256) void crf_gold_kernel(
    const float* __restrict__ scores, const int* __restrict__ targets,
    const int* __restrict__ lengths, float* __restrict__ wsGold)
{
    __shared__ float red[256];
    const int b = blockIdx.x, tid = threadIdx.x;
    const int len = lengths[b];
    float s = 0.0f;
    for (int t = tid; t < T_; t += 256)
        if (t < len)
            s += scores[(size_t)(b * T_ + t) * KK + targets[b * T_ + t]];
    red[tid] = s;
    __syncthreads();
    #pragma unroll
    for (int off = 128; off > 0; off >>= 1) {
        if (tid < off) red[tid] += red[tid + off];
        __syncthreads();
    }
    if (tid == 0) wsGold[b] = red[0];
}

// ---------------------------------------------------------------------------
// Final scalar: (sum_b alphaEnd - sum_b gold) / B    (one wave32)
// ---------------------------------------------------------------------------
__global__ void crf_final_kernel(const float* __restrict__ wsGold,
                                 const float* __restrict__ wsAlphaEnd,
                                 float* __restrict__ out)
{
    const int tid = threadIdx.x;  // 32 threads
    float d = wsAlphaEnd[tid] - wsGold[tid];
    #pragma unroll
    for (int off = 16; off > 0; off >>= 1) d += __shfl_down(d, off, 32);
    if (tid == 0) out[0] = d / (float)B_;
}

extern "C" void kernel_launch(void* const* d_in, const int* in_sizes, int n_in,
                              void* d_out, int out_size, void* d_ws, size_t ws_size,
                              hipStream_t stream) {
    const float* scores  = (const float*)d_in[0];
    const int*   targets = (const int*)d_in[1];
    const int*   lengths = (const int*)d_in[2];
    float*       out     = (float*)d_out;

    float* wsP        = (float*)d_ws;                       // B*C*48*48 floats
    float* wsSigma    = wsP + (size_t)B_ * CHUNKS * KK;     // B*C
    float* wsGold     = wsSigma + B_ * CHUNKS;              // B
    float* wsAlphaEnd = wsGold + B_;                        // B

    crf_chunk_kernel<<<dim3(B_ * CHUNKS), dim3(NTHR), 0, stream>>>(
        scores, lengths, wsP, wsSigma);
    crf_gold_kernel<<<dim3(B_), dim3(256), 0, stream>>>(
        scores, targets, lengths, wsGold);
    crf_combine_kernel<<<dim3(B_), dim3(64), 0, stream>>>(
        scores, wsP, wsSigma, wsAlphaEnd);
    crf_final_kernel<<<dim3(1), dim3(32), 0, stream>>>(
        wsGold, wsAlphaEnd, out);
}